// BeliefPropagationSingle_45792941310096
// MI455X (gfx1250) — compile-verified
//
#include <hip/hip_runtime.h>
#include <math.h>

// Problem dims (fixed by the reference)
#define D0   2048
#define D1F  8192
#define D1C  4096
#define D2   2048
#define N0   (D0 * D1F)   // 16777216  (t0 elements)
#define N1   (D1C * D2)   //  8388608  (t1 elements)

typedef __attribute__((ext_vector_type(2))) float v2f;
typedef __attribute__((ext_vector_type(8))) float v8f;

// ---- ordered-uint encoding of float for atomicMax (handles -inf correctly) ----
#define ENC_NEG_INF 0x007FFFFFu   // f2ord(-inf)

__device__ __forceinline__ unsigned f2ord(float f) {
  unsigned u = __float_as_uint(f);
  return (u & 0x80000000u) ? ~u : (u | 0x80000000u);
}
__device__ __forceinline__ float ord2f(unsigned u) {
  unsigned v = (u & 0x80000000u) ? (u & 0x7FFFFFFFu) : ~u;
  return __uint_as_float(v);
}

__device__ __forceinline__ float wredMax(float v) {
#pragma unroll
  for (int o = 16; o > 0; o >>= 1) v = fmaxf(v, __shfl_xor(v, o, 32));
  return v;
}
__device__ __forceinline__ float wredSum(float v) {
#pragma unroll
  for (int o = 16; o > 0; o >>= 1) v += __shfl_xor(v, o, 32);
  return v;
}

// ---------------- init atomic targets ----------------
__global__ void k_init(unsigned* msg_mx_u, float* msg_s, unsigned* scal_u) {
  int i = blockIdx.x * blockDim.x + threadIdx.x;
  if (i < D1C) { msg_mx_u[i] = ENC_NEG_INF; msg_s[i] = 0.0f; }
  if (i == 0) {
    scal_u[0] = ENC_NEG_INF;                 // g0max
    ((float*)scal_u)[1] = 0.0f;              // g0sum
    scal_u[2] = ENC_NEG_INF;                 // g1max
    ((float*)scal_u)[3] = 0.0f;              // g1sum
  }
}

// ---------------- marg_a[c] = LSE over rows of theta0 (column LSE) ----------------
// One wave per 16 columns. Sweep 1: column max. Sweep 2: sum of exp via
// V_WMMA_F32_16X16X4_F32 with A = ones(16x4), B = data(4x16) -> D[m][n] = colsum(n).
__global__ void k_marg_a(const float* __restrict__ theta0, float* __restrict__ marg_a) {
  const int wave = (blockIdx.x * blockDim.x + threadIdx.x) >> 5;
  const int lane = threadIdx.x & 31;
  const int n = lane & 15;       // column within group (== B-matrix N)
  const int h = lane >> 4;       // which K-pair this lane supplies
  const int c = (wave << 4) + n;

  // sweep 1: max over the 2048 rows of column c (split rows between lane pair)
  float m = -__builtin_inff();
  for (int t = 0; t < D0 / 4; ++t) {
    const int r = 4 * t + 2 * h;
    m = fmaxf(m, theta0[(size_t)r * D1F + c]);
    m = fmaxf(m, theta0[(size_t)(r + 1) * D1F + c]);
  }
  m = fmaxf(m, __shfl_xor(m, 16, 32));   // combine the two half-columns

  // sweep 2: sum of exp via WMMA (ones x data), fp32 exact
  v8f acc = {};
  v2f ones; ones.x = 1.0f; ones.y = 1.0f;
  for (int t = 0; t < D0 / 4; ++t) {
    const int r = 4 * t + 2 * h;
    v2f b;
    b.x = expf(theta0[(size_t)r * D1F + c] - m);
    b.y = expf(theta0[(size_t)(r + 1) * D1F + c] - m);
    acc = __builtin_amdgcn_wmma_f32_16x16x4_f32(false, ones, false, b,
                                                (short)0, acc, false, false);
  }
  // D[m][n] identical over m; VGPR0: lanes0-15 -> (M=0, N=lane)
  const float colsum = acc[0];
  if (lane < 16) marg_a[c] = m + logf(colsum);
}

// ---------------- segmented LSE of marg_a into msg_ab (idx_b) ----------------
__global__ void k_seg_max(const float* __restrict__ marg_a, const int* __restrict__ idx_b,
                          unsigned* __restrict__ msg_mx_u) {
  int i = blockIdx.x * blockDim.x + threadIdx.x;
  if (i < D1F) atomicMax(&msg_mx_u[idx_b[i]], f2ord(marg_a[i]));
}
__global__ void k_seg_sum(const float* __restrict__ marg_a, const int* __restrict__ idx_b,
                          const unsigned* __restrict__ msg_mx_u, float* __restrict__ msg_s) {
  int i = blockIdx.x * blockDim.x + threadIdx.x;
  if (i < D1F) {
    int seg = idx_b[i];
    float sh = marg_a[i] - ord2f(msg_mx_u[seg]);
    if (sh != sh) sh = 0.0f;                       // nan_to_num (reference semantics)
    atomicAdd(&msg_s[seg], expf(sh));
  }
}
__global__ void k_seg_fin(const unsigned* __restrict__ msg_mx_u, const float* __restrict__ msg_s,
                          float* __restrict__ msg_ab) {
  int i = blockIdx.x * blockDim.x + threadIdx.x;
  if (i < D1C) msg_ab[i] = logf(msg_s[i]) + ord2f(msg_mx_u[i]);  // empty seg -> -inf
}

// ---------------- marg_b[r] = LSE over cols of theta1 (row LSE) ----------------
// One wave per 16 rows. Sweep 2 uses WMMA with A = data(16x4), B = ones(4x16)
// -> D[m][n] = rowsum(m) for all n.
__global__ void k_marg_b(const float* __restrict__ theta1, float* __restrict__ marg_b) {
  const int wave = (blockIdx.x * blockDim.x + threadIdx.x) >> 5;
  const int lane = threadIdx.x & 31;
  const int n = lane & 15;       // row within group (== A-matrix M)
  const int h = lane >> 4;       // which K-pair this lane supplies
  const int row = (wave << 4) + n;
  const float* rp = theta1 + (size_t)row * D2;

  float m = -__builtin_inff();
  for (int t = 0; t < D2 / 4; ++t) {
    const int cb = 4 * t + 2 * h;
    m = fmaxf(m, rp[cb]);
    m = fmaxf(m, rp[cb + 1]);
  }
  m = fmaxf(m, __shfl_xor(m, 16, 32));

  v8f acc = {};
  v2f ones; ones.x = 1.0f; ones.y = 1.0f;
  for (int t = 0; t < D2 / 4; ++t) {
    const int cb = 4 * t + 2 * h;
    v2f a;
    a.x = expf(rp[cb] - m);
    a.y = expf(rp[cb + 1] - m);
    acc = __builtin_amdgcn_wmma_f32_16x16x4_f32(false, a, false, ones,
                                                (short)0, acc, false, false);
  }
  // Extract rowsum(row) from the C/D layout:
  //   m<8 : VGPR m,   lanes 0-15  hold (M=m, N=lane)
  //   m>=8: VGPR m-8, lanes 16-31 hold (M=m, N=lane-16)
  const int ri = n & 7;
  float x = acc[0];
#pragma unroll
  for (int i = 1; i < 8; ++i) x = (ri == i) ? acc[i] : x;
  const float rowsum = __shfl(x, n + (n & 8), 32);
  if (lane < 16) marg_b[row] = m + logf(rowsum);
}

// ---------------- msg_ba[i] = marg_b[idx_a[i]] ----------------
__global__ void k_gather(const float* __restrict__ marg_b, const int* __restrict__ idx_a,
                         float* __restrict__ msg_ba) {
  int i = blockIdx.x * blockDim.x + threadIdx.x;
  if (i < D1F) msg_ba[i] = marg_b[idx_a[i]];
}

// ---------------- t0 / t1: build, global max, global sum, normalize ----------------
__global__ void k_t0_max(const float* __restrict__ theta0, const float* __restrict__ msg_ba,
                         float* __restrict__ out0, unsigned* __restrict__ gmax) {
  const size_t stride = (size_t)gridDim.x * blockDim.x;
  float m = -__builtin_inff();
  for (size_t i = blockIdx.x * (size_t)blockDim.x + threadIdx.x; i < (size_t)N0; i += stride) {
    float x = theta0[i] + msg_ba[i & (D1F - 1)];
    out0[i] = x;
    m = fmaxf(m, x);
  }
  m = wredMax(m);
  if ((threadIdx.x & 31) == 0) atomicMax(gmax, f2ord(m));
}
__global__ void k_t1_max(const float* __restrict__ theta1, const float* __restrict__ msg_ab,
                         float* __restrict__ out1, unsigned* __restrict__ gmax) {
  const size_t stride = (size_t)gridDim.x * blockDim.x;
  float m = -__builtin_inff();
  for (size_t i = blockIdx.x * (size_t)blockDim.x + threadIdx.x; i < (size_t)N1; i += stride) {
    float x = theta1[i] + msg_ab[i >> 11];   // D2 == 2048
    out1[i] = x;
    m = fmaxf(m, x);
  }
  m = wredMax(m);
  if ((threadIdx.x & 31) == 0) atomicMax(gmax, f2ord(m));
}
__global__ void k_sumexp(const float* __restrict__ out, size_t n,
                         const unsigned* __restrict__ gmax, float* __restrict__ gsum) {
  const float M = ord2f(*gmax);
  const size_t stride = (size_t)gridDim.x * blockDim.x;
  float s = 0.0f;
  for (size_t i = blockIdx.x * (size_t)blockDim.x + threadIdx.x; i < n; i += stride)
    s += expf(out[i] - M);
  s = wredSum(s);
  if ((threadIdx.x & 31) == 0) atomicAdd(gsum, s);
}
__global__ void k_norm(float* __restrict__ out, size_t n,
                       const unsigned* __restrict__ gmax, const float* __restrict__ gsum) {
  const float Z = ord2f(*gmax) + logf(*gsum);
  const size_t stride = (size_t)gridDim.x * blockDim.x;
  for (size_t i = blockIdx.x * (size_t)blockDim.x + threadIdx.x; i < n; i += stride)
    out[i] -= Z;
}

extern "C" void kernel_launch(void* const* d_in, const int* in_sizes, int n_in,
                              void* d_out, int out_size, void* d_ws, size_t ws_size,
                              hipStream_t stream) {
  const float* theta0 = (const float*)d_in[0];
  const float* theta1 = (const float*)d_in[1];
  const int*   idx_a  = (const int*)d_in[2];
  const int*   idx_b  = (const int*)d_in[3];

  float* out0 = (float*)d_out;          // D0 x D1F
  float* out1 = out0 + (size_t)N0;      // D1C x D2

  // workspace layout (floats)
  float*    ws       = (float*)d_ws;
  float*    marg_a   = ws;                         // 8192
  unsigned* msg_mx_u = (unsigned*)(ws + 8192);     // 4096
  float*    msg_s    = ws + 12288;                 // 4096
  float*    msg_ab   = ws + 16384;                 // 4096
  float*    marg_b   = ws + 20480;                 // 4096
  float*    msg_ba   = ws + 24576;                 // 8192
  unsigned* scal     = (unsigned*)(ws + 32768);    // [g0max_u, g0sum_f, g1max_u, g1sum_f]
  unsigned* g0max = scal + 0;  float* g0sum = (float*)(scal + 1);
  unsigned* g1max = scal + 2;  float* g1sum = (float*)(scal + 3);

  const int B = 256;

  k_init<<<(D1C + B - 1) / B, B, 0, stream>>>(msg_mx_u, msg_s, scal);

  // column LSE of theta0 : 8192/16 = 512 waves = 64 blocks of 256
  k_marg_a<<<64, B, 0, stream>>>(theta0, marg_a);

  k_seg_max<<<D1F / B, B, 0, stream>>>(marg_a, idx_b, msg_mx_u);
  k_seg_sum<<<D1F / B, B, 0, stream>>>(marg_a, idx_b, msg_mx_u, msg_s);
  k_seg_fin<<<D1C / B, B, 0, stream>>>(msg_mx_u, msg_s, msg_ab);

  // row LSE of theta1 : 4096/16 = 256 waves = 32 blocks of 256
  k_marg_b<<<32, B, 0, stream>>>(theta1, marg_b);
  k_gather<<<D1F / B, B, 0, stream>>>(marg_b, idx_a, msg_ba);

  k_t0_max<<<4096, B, 0, stream>>>(theta0, msg_ba, out0, g0max);
  k_t1_max<<<2048, B, 0, stream>>>(theta1, msg_ab, out1, g1max);
  k_sumexp<<<4096, B, 0, stream>>>(out0, (size_t)N0, g0max, g0sum);
  k_sumexp<<<2048, B, 0, stream>>>(out1, (size_t)N1, g1max, g1sum);
  k_norm<<<4096, B, 0, stream>>>(out0, (size_t)N0, g0max, g0sum);
  k_norm<<<2048, B, 0, stream>>>(out1, (size_t)N1, g1max, g1sum);
}